// HGNConv_70592082477284
// MI455X (gfx1250) — compile-verified
//
#include <hip/hip_runtime.h>
#include <hip/hip_bf16.h>

#define H 8
#define F 32
#define D 32
#define HF 256
#define DIN 256
#define N0_SPLIT 1373
#define TAO 0.4f
#define EPSV 1e-12f

typedef float v2f __attribute__((ext_vector_type(2)));
typedef float v8f __attribute__((ext_vector_type(8)));

// ---------------------------------------------------------------------------
// GEMM: Y[row_start:row_end, 0:256] = X[row_start:row_end, 0:256] @ W[256,256]
// One block = 256 threads = 8 waves. Block covers a 16-row strip; A-tile is
// staged in LDS (stride 260 floats -> conflict-free 16-row walk). Each wave
// owns a 16x32 output slab (two adjacent 16-col WMMA tiles) so every A
// fragment loaded from LDS feeds TWO V_WMMA_F32_16X16X4_F32 issues.
// EXEC stays all-ones (WMMA requirement); boundary tiles clamp loads and take
// a masked store path, interior tiles store unconditionally (uniform branch).
// ---------------------------------------------------------------------------
#define ASTRIDE 260

__global__ __launch_bounds__(256) void gemm_wmma_f32(
    const float* __restrict__ X, const float* __restrict__ W,
    float* __restrict__ Y, int row_start, int row_end) {
  __shared__ float Atile[16 * ASTRIDE];
  const int m0  = row_start + blockIdx.x * 16;
  const int tid = threadIdx.x;

  // Cooperative, coalesced A-tile load (rows clamped so we never read OOB).
  for (int idx = tid; idx < 16 * DIN; idx += 256) {
    int r = m0 + (idx >> 8);
    int c = idx & 255;
    int rr = r < row_end ? r : (row_end - 1);
    Atile[(idx >> 8) * ASTRIDE + c] = X[(size_t)rr * DIN + c];
  }
  __syncthreads();

  const int wave = tid >> 5;
  const int lane = tid & 31;
  const int half = lane >> 4;   // K sub-pair selector (ISA A/B layout)
  const int lm   = lane & 15;   // M (for A) / N (for B) index
  const int col0 = wave * 32;   // two adjacent 16-col tiles: col0, col0+16

  v8f acc0 = {}, acc1 = {};
#if __has_builtin(__builtin_amdgcn_wmma_f32_16x16x4_f32)
#pragma unroll 8
  for (int k = 0; k < DIN; k += 4) {
    const int ka = k + half * 2;
    v2f a, b0, b1;
    a.x = Atile[lm * ASTRIDE + ka];
    a.y = Atile[lm * ASTRIDE + ka + 1];
    const float* wr = W + (size_t)ka * HF + col0 + lm;
    b0.x = wr[0];       b0.y = wr[HF];
    b1.x = wr[16];      b1.y = wr[16 + HF];
    acc0 = __builtin_amdgcn_wmma_f32_16x16x4_f32(
        false, a, false, b0, (short)0, acc0, false, false);
    acc1 = __builtin_amdgcn_wmma_f32_16x16x4_f32(
        false, a, false, b1, (short)0, acc1, false, false);
  }
#else  // scalar fallback (should not be taken on gfx1250)
#pragma unroll
  for (int r = 0; r < 8; ++r) {
    float s0 = 0.f, s1 = 0.f;
    int row = half * 8 + r;
    for (int k = 0; k < DIN; ++k) {
      float av = Atile[row * ASTRIDE + k];
      s0 = fmaf(av, W[(size_t)k * HF + col0 + lm], s0);
      s1 = fmaf(av, W[(size_t)k * HF + col0 + 16 + lm], s1);
    }
    acc0[r] = s0; acc1[r] = s1;
  }
#endif

  const int rbase = m0 + half * 8;   // ISA C/D layout: M = r (+8 for hi half)
  if (m0 + 16 <= row_end) {          // wave-uniform fast path: full tile
#pragma unroll
    for (int r = 0; r < 8; ++r) {
      float* yr = Y + (size_t)(rbase + r) * HF + lm;
      yr[col0]      = acc0[r];
      yr[col0 + 16] = acc1[r];
    }
  } else {                           // boundary tile: mask rows
#pragma unroll
    for (int r = 0; r < 8; ++r) {
      if (rbase + r < row_end) {
        float* yr = Y + (size_t)(rbase + r) * HF + lm;
        yr[col0]      = acc0[r];
        yr[col0 + 16] = acc1[r];
      }
    }
  }
}

// ---------------------------------------------------------------------------
// ft_attn = softmax_g(tanh(feat_src[n,h,:] @ W_prob[h]))
// h is wave-uniform -> W_prob reads become scalar (SMEM) loads.
// ---------------------------------------------------------------------------
__global__ __launch_bounds__(256) void attn_prob_kernel(
    const float* __restrict__ fsrc, const float* __restrict__ Wp,
    float* __restrict__ fattn, int N) {
  const int h = threadIdx.x >> 5;                  // uniform per wave
  const int n = blockIdx.x * 32 + (threadIdx.x & 31);
  if (n >= N) return;
  float x[F];
#pragma unroll
  for (int f = 0; f < F; ++f) x[f] = fsrc[(size_t)n * HF + h * F + f];
  float y[F];
  float mx = -3.0e38f;
#pragma unroll 4
  for (int g = 0; g < F; ++g) {
    float s = 0.f;
#pragma unroll
    for (int f = 0; f < F; ++f)
      s = fmaf(x[f], Wp[(size_t)h * F * F + f * F + g], s);
    y[g] = tanhf(s);
    mx = fmaxf(mx, y[g]);
  }
  float z = 0.f;
#pragma unroll
  for (int g = 0; g < F; ++g) { y[g] = expf(y[g] - mx); z += y[g]; }
  const float inv = 1.f / z;
#pragma unroll
  for (int g = 0; g < F; ++g)
    fattn[(size_t)n * HF + h * F + g] = y[g] * inv;
}

// ---------------------------------------------------------------------------
// Column sum-of-squares of feat_src (l2norm over axis 0), then reciprocal.
// ---------------------------------------------------------------------------
__global__ __launch_bounds__(256) void colsq_kernel(
    const float* __restrict__ fsrc, float* __restrict__ colss,
    int N, int rows_per_block) {
  const int c = threadIdx.x;
  const int r0 = blockIdx.x * rows_per_block;
  const int r1 = min(r0 + rows_per_block, N);
  float s = 0.f;
  for (int r = r0; r < r1; ++r) {
    float v = fsrc[(size_t)r * HF + c];
    s = fmaf(v, v, s);
  }
  atomicAdd(&colss[c], s);
}

__global__ void colinv_kernel(const float* __restrict__ colss,
                              float* __restrict__ invn) {
  int c = threadIdx.x;
  invn[c] = 1.f / fmaxf(sqrtf(colss[c]), EPSV);
}

// ---------------------------------------------------------------------------
// Row-wise L2 normalize 32-d vectors (sloc / topo). One wave per node.
// ---------------------------------------------------------------------------
__device__ __forceinline__ float wave_sum(float v) {
#pragma unroll
  for (int m = 16; m >= 1; m >>= 1) v += __shfl_xor(v, m, 32);
  return v;
}

__global__ __launch_bounds__(256) void vecnorm_kernel(
    const float* __restrict__ x, float* __restrict__ y, int N) {
  const int n = blockIdx.x * 8 + (threadIdx.x >> 5);
  if (n >= N) return;
  const int i = threadIdx.x & 31;
  float v = x[(size_t)n * D + i];
  float ss = wave_sum(v * v);
  y[(size_t)n * D + i] = v / fmaxf(sqrtf(ss), EPSV);
}

// ---------------------------------------------------------------------------
// Edge scores. One wave per edge. Lane i covers (h=k, f=i) for k=0..7.
// ---------------------------------------------------------------------------
__global__ __launch_bounds__(256) void edge_score_kernel(
    const float* __restrict__ fsrc, const float* __restrict__ fattn,
    const float* __restrict__ invn, const float* __restrict__ slocn,
    const float* __restrict__ topon, const int* __restrict__ src,
    const int* __restrict__ dst, const int* __restrict__ efeat,
    const float* __restrict__ etype, float* __restrict__ esc, int E) {
  const int e = blockIdx.x * 8 + (threadIdx.x >> 5);
  if (e >= E) return;
  const int i = threadIdx.x & 31;
  const int s = src[e], d = dst[e];
  float a[H];
#pragma unroll
  for (int k = 0; k < H; ++k) {
    const int c = k * F + i;
    a[k] = wave_sum(fattn[(size_t)d * HF + c] * fsrc[(size_t)s * HF + c] * invn[c]);
  }
  const float cs = wave_sum(slocn[(size_t)s * D + i] * slocn[(size_t)d * D + i]);
  const float ct = wave_sum(topon[(size_t)s * D + i] * topon[(size_t)d * D + i]);
  const float sim = (TAO * cs + (1.f - TAO) * ct + 1.f) * 0.5f;
  const float w = etype[efeat[e]];
  if (i == 0) {
#pragma unroll
    for (int k = 0; k < H; ++k) esc[(size_t)e * H + k] = sim * a[k] * w;
  }
}

// ---------------------------------------------------------------------------
// Segment softmax over dst (atomic max, then exp + atomic sum).
// ---------------------------------------------------------------------------
__device__ __forceinline__ void atomicMaxF(float* addr, float val) {
  if (val >= 0.f) atomicMax((int*)addr, __float_as_int(val));
  else            atomicMin((unsigned int*)addr, __float_as_uint(val));
}

__global__ void fill_kernel(float* __restrict__ p, float v, size_t n) {
  size_t t = (size_t)blockIdx.x * blockDim.x + threadIdx.x;
  if (t < n) p[t] = v;
}

__global__ void seg_max_kernel(const float* __restrict__ esc,
                               const int* __restrict__ dst,
                               float* __restrict__ m, int E) {
  int t = blockIdx.x * blockDim.x + threadIdx.x;
  if (t >= E * H) return;
  int d = dst[t >> 3];
  atomicMaxF(&m[(size_t)d * H + (t & 7)], esc[t]);
}

__global__ void seg_expsum_kernel(float* __restrict__ esc,
                                  const int* __restrict__ dst,
                                  const float* __restrict__ m,
                                  float* __restrict__ z, int E) {
  int t = blockIdx.x * blockDim.x + threadIdx.x;
  if (t >= E * H) return;
  int d = dst[t >> 3];
  float v = expf(esc[t] - m[(size_t)d * H + (t & 7)]);
  esc[t] = v;                       // in-place: esc now holds ex
  atomicAdd(&z[(size_t)d * H + (t & 7)], v);
}

// ---------------------------------------------------------------------------
// Weighted aggregation. One wave per edge; lane i handles f=i for all heads.
// ---------------------------------------------------------------------------
__global__ __launch_bounds__(256) void aggregate_kernel(
    const float* __restrict__ ex, const float* __restrict__ z,
    const float* __restrict__ fsrc, const float* __restrict__ sloc,
    const float* __restrict__ topo, const int* __restrict__ src,
    const int* __restrict__ dst, float* __restrict__ rst,
    float* __restrict__ accS, float* __restrict__ accT, int E) {
  const int e = blockIdx.x * 8 + (threadIdx.x >> 5);
  if (e >= E) return;
  const int i = threadIdx.x & 31;
  const int s = src[e], d = dst[e];
  float att[H];
  float msum = 0.f;
#pragma unroll
  for (int k = 0; k < H; ++k) {
    att[k] = ex[(size_t)e * H + k] / z[(size_t)d * H + k];
    msum += att[k];
  }
  const float matt = msum * (1.f / H);
#pragma unroll
  for (int k = 0; k < H; ++k)
    atomicAdd(&rst[(size_t)d * HF + k * F + i],
              att[k] * fsrc[(size_t)s * HF + k * F + i]);
  atomicAdd(&accS[(size_t)d * D + i], matt * sloc[(size_t)s * D + i]);
  atomicAdd(&accT[(size_t)d * D + i], matt * topo[(size_t)s * D + i]);
}

// ---------------------------------------------------------------------------
// Finalize: out_fea = mean_h relu(rst+resval); out_sloc/topo = acc + input.
// ---------------------------------------------------------------------------
__global__ void finalize_kernel(const float* __restrict__ rst,
                                const float* __restrict__ resv,
                                const float* __restrict__ accS,
                                const float* __restrict__ accT,
                                const float* __restrict__ sloc,
                                const float* __restrict__ topo,
                                float* __restrict__ out, int N) {
  int t = blockIdx.x * blockDim.x + threadIdx.x;
  if (t >= N * D) return;
  int n = t >> 5, i = t & 31;
  float s = 0.f;
#pragma unroll
  for (int h = 0; h < H; ++h) {
    float v = rst[(size_t)n * HF + h * F + i] + resv[(size_t)n * HF + h * F + i];
    s += fmaxf(v, 0.f);
  }
  const size_t nd = (size_t)N * D;
  out[t]          = s * (1.f / H);
  out[nd + t]     = accS[t] + sloc[t];
  out[2 * nd + t] = accT[t] + topo[t];
}

// ---------------------------------------------------------------------------

extern "C" void kernel_launch(void* const* d_in, const int* in_sizes, int n_in,
                              void* d_out, int out_size, void* d_ws, size_t ws_size,
                              hipStream_t stream) {
  const float* feat  = (const float*)d_in[0];
  const float* sloc  = (const float*)d_in[1];
  const float* topo  = (const float*)d_in[2];
  const int*   src   = (const int*)d_in[3];
  const int*   dst   = (const int*)d_in[4];
  const int*   efeat = (const int*)d_in[5];
  const float* Wp0   = (const float*)d_in[6];
  const float* Wp1   = (const float*)d_in[7];
  const float* Wprob = (const float*)d_in[8];
  const float* Wr0   = (const float*)d_in[9];
  const float* Wr1   = (const float*)d_in[10];
  const float* etype = (const float*)d_in[11];

  const int N = in_sizes[0] / DIN;
  const int E = in_sizes[3];
  const int n0 = (N0_SPLIT < N) ? N0_SPLIT : N;

  float* ws = (float*)d_ws;
  float* fsrc  = ws; ws += (size_t)N * HF;
  float* resv  = ws; ws += (size_t)N * HF;
  float* fattn = ws; ws += (size_t)N * HF;
  float* rst   = ws; ws += (size_t)N * HF;
  float* slocn = ws; ws += (size_t)N * D;
  float* topon = ws; ws += (size_t)N * D;
  float* accS  = ws; ws += (size_t)N * D;
  float* accT  = ws; ws += (size_t)N * D;
  float* mbuf  = ws; ws += (size_t)N * H;
  float* zbuf  = ws; ws += (size_t)N * H;
  float* colss = ws; ws += 256;
  float* invn  = ws; ws += 256;
  float* esc   = ws; ws += (size_t)E * H;

  // Zero accumulators (memset nodes are graph-capture safe).
  hipMemsetAsync(rst,   0, (size_t)N * HF * sizeof(float), stream);
  hipMemsetAsync(accS,  0, (size_t)N * D * sizeof(float), stream);
  hipMemsetAsync(accT,  0, (size_t)N * D * sizeof(float), stream);
  hipMemsetAsync(zbuf,  0, (size_t)N * H * sizeof(float), stream);
  hipMemsetAsync(colss, 0, 256 * sizeof(float), stream);
  {
    size_t n = (size_t)N * H;
    fill_kernel<<<(unsigned)((n + 255) / 256), 256, 0, stream>>>(mbuf, -3.0e38f, n);
  }

  // Dense projections (WMMA f32), split at N0 per-type weights.
  const int t0 = (n0 + 15) / 16;
  const int t1 = (N - n0 + 15) / 16;
  if (t0 > 0) {
    gemm_wmma_f32<<<t0, 256, 0, stream>>>(feat, Wp0, fsrc, 0, n0);
    gemm_wmma_f32<<<t0, 256, 0, stream>>>(feat, Wr0, resv, 0, n0);
  }
  if (t1 > 0) {
    gemm_wmma_f32<<<t1, 256, 0, stream>>>(feat, Wp1, fsrc, n0, N);
    gemm_wmma_f32<<<t1, 256, 0, stream>>>(feat, Wr1, resv, n0, N);
  }

  // Per-head probability softmax + norms.
  attn_prob_kernel<<<(N + 31) / 32, 256, 0, stream>>>(fsrc, Wprob, fattn, N);
  {
    const int nblk = 100;
    const int rpb = (N + nblk - 1) / nblk;
    colsq_kernel<<<nblk, 256, 0, stream>>>(fsrc, colss, N, rpb);
    colinv_kernel<<<1, 256, 0, stream>>>(colss, invn);
  }
  vecnorm_kernel<<<(N + 7) / 8, 256, 0, stream>>>(sloc, slocn, N);
  vecnorm_kernel<<<(N + 7) / 8, 256, 0, stream>>>(topo, topon, N);

  // Edge scores -> segment softmax -> aggregation.
  const int eblk = (E + 7) / 8;
  edge_score_kernel<<<eblk, 256, 0, stream>>>(fsrc, fattn, invn, slocn, topon,
                                              src, dst, efeat, etype, esc, E);
  const int ehblk = (E * H + 255) / 256;
  seg_max_kernel<<<ehblk, 256, 0, stream>>>(esc, dst, mbuf, E);
  seg_expsum_kernel<<<ehblk, 256, 0, stream>>>(esc, dst, mbuf, zbuf, E);
  aggregate_kernel<<<eblk, 256, 0, stream>>>(esc, zbuf, fsrc, sloc, topo,
                                             src, dst, rst, accS, accT, E);

  finalize_kernel<<<(N * D + 255) / 256, 256, 0, stream>>>(
      rst, resv, accS, accT, sloc, topo, (float*)d_out, N);
}